// SelfAttentionLayerMdf_1563368095895
// MI455X (gfx1250) — compile-verified
//
#include <hip/hip_runtime.h>

// Problem constants (from reference): B=8, N=2048, C=256, W=256
#define B_ 8
#define N_ 2048
#define C_ 256
#define W_ 256

typedef float v2f __attribute__((ext_vector_type(2)));
typedef float v8f __attribute__((ext_vector_type(8)));

// CDNA5 fp32 WMMA: D(16x16,f32) = A(16x4,f32) * B(4x16,f32) + C
// 8-arg form: (neg_a, A, neg_b, B, c_mod, C, reuse_a, reuse_b)
__device__ __forceinline__ v8f wmma_f32_16x16x4(v2f a, v2f b, v8f c) {
  return __builtin_amdgcn_wmma_f32_16x16x4_f32(false, a, false, b, (short)0, c,
                                               false, false);
}

// ---------------------------------------------------------------------------
// Kernel 1: Q/K/V = relu(X * W^T + bias)
//   X: [B*N, C] row-major, Wm: [W, C] row-major (so B-matrix B[c][w] = Wm[w*C+c],
//   i.e. a contiguous-in-c read -> float2 loads for both A and B fragments).
//   grid.x over (M/16)*(W/16)/8 tiles, grid.y selects Q/K/V.
// ---------------------------------------------------------------------------
__global__ void qkv_kernel(const float* __restrict__ x,
                           const float* __restrict__ Wq, const float* __restrict__ bq,
                           const float* __restrict__ Wk, const float* __restrict__ bk,
                           const float* __restrict__ Wv, const float* __restrict__ bv,
                           float* __restrict__ Q, float* __restrict__ K,
                           float* __restrict__ V) {
  const float* Wm;
  const float* bias;
  float* out;
  if (blockIdx.y == 0) { Wm = Wq; bias = bq; out = Q; }
  else if (blockIdx.y == 1) { Wm = Wk; bias = bk; out = K; }
  else { Wm = Wv; bias = bv; out = V; }

  const int wave = threadIdx.x >> 5;
  const int lane = threadIdx.x & 31;
  const int half = lane >> 4;   // which half of the wave
  const int r    = lane & 15;   // row/col within tile

  const int tiles_n = W_ / 16;                 // 16
  const int tile = blockIdx.x * 8 + wave;      // 0 .. (M/16)*(W/16)-1
  const int m0 = (tile / tiles_n) * 16;        // over M = B*N
  const int n0 = (tile % tiles_n) * 16;

  const float* arow = x  + (size_t)(m0 + r) * C_;
  const float* brow = Wm + (size_t)(n0 + r) * C_;

  v8f acc = {};
#pragma unroll 4
  for (int k0 = 0; k0 < C_; k0 += 4) {
    v2f a = *(const v2f*)(arow + k0 + 2 * half);
    v2f b = *(const v2f*)(brow + k0 + 2 * half);
    acc = wmma_f32_16x16x4(a, b, acc);
  }

  const float bv_ = bias[n0 + r];
#pragma unroll
  for (int g = 0; g < 8; ++g) {
    const int m = m0 + g + 8 * half;
    float v = acc[g] + bv_;
    v = v > 0.0f ? v : 0.0f;                   // ReLU
    out[(size_t)m * W_ + n0 + r] = v;
  }
}

// ---------------------------------------------------------------------------
// Kernel 2: S[b,i,j] = (Q[b,i,:] . K[b,j,:]) / sqrt(N)
//   A[i][w] = Q row-major, B[w][j] = K[b,j,w] (contiguous in w -> float2 loads)
// ---------------------------------------------------------------------------
__global__ void scores_kernel(const float* __restrict__ Q,
                              const float* __restrict__ K,
                              float* __restrict__ S) {
  const int b    = blockIdx.y;
  const int wave = threadIdx.x >> 5;
  const int lane = threadIdx.x & 31;
  const int half = lane >> 4;
  const int r    = lane & 15;

  const int tiles_n = N_ / 16;                 // 128
  const int tile = blockIdx.x * 8 + wave;
  const int i0 = (tile / tiles_n) * 16;
  const int j0 = (tile % tiles_n) * 16;

  const float* arow = Q + ((size_t)b * N_ + i0 + r) * W_;
  const float* brow = K + ((size_t)b * N_ + j0 + r) * W_;

  v8f acc = {};
#pragma unroll 4
  for (int k0 = 0; k0 < W_; k0 += 4) {
    v2f a = *(const v2f*)(arow + k0 + 2 * half);
    v2f b2 = *(const v2f*)(brow + k0 + 2 * half);
    acc = wmma_f32_16x16x4(a, b2, acc);
  }

  const float scale = 0.022097086912079608f;   // 1/sqrt(2048)
  float* sb = S + (size_t)b * N_ * N_;
#pragma unroll
  for (int g = 0; g < 8; ++g) {
    const int i = i0 + g + 8 * half;
    sb[(size_t)i * N_ + j0 + r] = acc[g] * scale;
  }
}

// ---------------------------------------------------------------------------
// Kernel 3: softmax over the QUERY axis i (axis=1), i.e. per column (b,j):
//   m_j = max_i S ; S <- exp(S - m_j) in place ; Dsum[b,j] = sum_i exp.
//   One thread per column; adjacent threads read adjacent j -> fully coalesced.
// ---------------------------------------------------------------------------
__global__ void colsoftmax_kernel(float* __restrict__ S, float* __restrict__ Dsum) {
  const int t = blockIdx.x * blockDim.x + threadIdx.x;  // 0 .. B*N-1
  const int b = t / N_;
  const int j = t % N_;
  float* col = S + (size_t)b * N_ * N_ + j;

  float m = -3.4e38f;
#pragma unroll 4
  for (int i = 0; i < N_; ++i) {
    m = fmaxf(m, col[(size_t)i * N_]);
  }
  float d = 0.0f;
#pragma unroll 4
  for (int i = 0; i < N_; ++i) {
    const float e = __expf(col[(size_t)i * N_] - m);
    col[(size_t)i * N_] = e;
    d += e;
  }
  Dsum[t] = d;
}

// ---------------------------------------------------------------------------
// Kernel 4: fold the softmax normalizer into V:  V'[b,j,w] = V[b,j,w] / d_j
//   (1/d_j depends only on j == the K-dim of attn@V, so row-scaling V is exact
//    and avoids a third full pass over the 134MB score buffer)
// ---------------------------------------------------------------------------
__global__ void scalev_kernel(float* __restrict__ V, const float* __restrict__ Dsum) {
  const int idx = blockIdx.x * blockDim.x + threadIdx.x;  // float4 granularity
  const int total = B_ * N_ * W_ / 4;
  if (idx >= total) return;
  float4* v4 = (float4*)V;
  const int row = (idx * 4) / W_;                         // b*N + j
  const float inv = 1.0f / Dsum[row];
  float4 v = v4[idx];
  v.x *= inv; v.y *= inv; v.z *= inv; v.w *= inv;
  v4[idx] = v;
}

// ---------------------------------------------------------------------------
// Kernel 5: out[b,i,w] = sum_j P[b,i,j] * V'[b,j,w]  + Q[b,i,w]
//   A = P (contiguous in j -> float2), B[k=j][n=w] = V'[j*W + w] (two scalar
//   loads, lanes r=0..15 coalesce along w).
// ---------------------------------------------------------------------------
__global__ void av_kernel(const float* __restrict__ P, const float* __restrict__ V,
                          const float* __restrict__ Q, float* __restrict__ out) {
  const int b    = blockIdx.y;
  const int wave = threadIdx.x >> 5;
  const int lane = threadIdx.x & 31;
  const int half = lane >> 4;
  const int r    = lane & 15;

  const int tiles_n = W_ / 16;                 // 16
  const int tile = blockIdx.x * 8 + wave;
  const int i0 = (tile / tiles_n) * 16;
  const int n0 = (tile % tiles_n) * 16;

  const float* arow = P + (size_t)b * N_ * N_ + (size_t)(i0 + r) * N_;
  const float* vb   = V + (size_t)b * N_ * W_;

  v8f acc = {};
#pragma unroll 2
  for (int k0 = 0; k0 < N_; k0 += 4) {
    v2f a = *(const v2f*)(arow + k0 + 2 * half);
    const int kr = k0 + 2 * half;
    v2f b2;
    b2.x = vb[(size_t)kr * W_ + n0 + r];
    b2.y = vb[(size_t)(kr + 1) * W_ + n0 + r];
    acc = wmma_f32_16x16x4(a, b2, acc);
  }

  const float* qb = Q + (size_t)b * N_ * W_;
  float* ob = out + (size_t)b * N_ * W_;
#pragma unroll
  for (int g = 0; g < 8; ++g) {
    const int i = i0 + g + 8 * half;
    ob[(size_t)i * W_ + n0 + r] = acc[g] + qb[(size_t)i * W_ + n0 + r];
  }
}

// ---------------------------------------------------------------------------
extern "C" void kernel_launch(void* const* d_in, const int* in_sizes, int n_in,
                              void* d_out, int out_size, void* d_ws, size_t ws_size,
                              hipStream_t stream) {
  const float* x  = (const float*)d_in[0];
  const float* Wq = (const float*)d_in[1];
  const float* bq = (const float*)d_in[2];
  const float* Wk = (const float*)d_in[3];
  const float* bk = (const float*)d_in[4];
  const float* Wv = (const float*)d_in[5];
  const float* bv = (const float*)d_in[6];
  // d_in[7] = valid_len: unused in the reference forward.
  float* out = (float*)d_out;
  float* ws  = (float*)d_ws;

  const size_t QKV = (size_t)B_ * N_ * W_;     // 4,194,304 floats
  float* Q = ws;
  float* K = ws + QKV;
  float* V = ws + 2 * QKV;
  float* D = ws + 3 * QKV;                     // B*N colsum
  float* S = D + (size_t)B_ * N_;              // B*N*N scores / probs

  // 1) Q,K,V = relu(X W^T + b)
  {
    dim3 grid(((B_ * N_ / 16) * (W_ / 16)) / 8, 3);
    qkv_kernel<<<grid, 256, 0, stream>>>(x, Wq, bq, Wk, bk, Wv, bv, Q, K, V);
  }
  // 2) S = Q K^T / sqrt(N)
  {
    dim3 grid(((N_ / 16) * (N_ / 16)) / 8, B_);
    scores_kernel<<<grid, 256, 0, stream>>>(Q, K, S);
  }
  // 3) column softmax stats: S <- exp(S - m_j), D = colsum
  colsoftmax_kernel<<<dim3((B_ * N_) / 256), 256, 0, stream>>>(S, D);
  // 4) V <- V / d_j (fold normalizer into V rows)
  scalev_kernel<<<dim3((B_ * N_ * W_ / 4) / 256), 256, 0, stream>>>(V, D);
  // 5) out = P V' + Q
  {
    dim3 grid(((N_ / 16) * (W_ / 16)) / 8, B_);
    av_kernel<<<grid, 256, 0, stream>>>(S, V, Q, out);
  }
}